// CARAFEPack_40389872452089
// MI455X (gfx1250) — compile-verified
//
#include <hip/hip_runtime.h>
#include <math.h>

// Problem constants (from reference)
#define NN 4
#define CI 256
#define HH 64
#define WW 64
#define HWSZ (HH*WW)       // 4096
#define CCOMP 64
#define MC 100             // K*K*G*S*S
#define MCP 112            // padded to 7 x 16 for WMMA M tiles
#define KK9 576            // 64 * 3 * 3 im2col K
#define KTAPS 25
#define HO (HH*2)
#define WO (WW*2)

typedef float v2f __attribute__((ext_vector_type(2)));
typedef float v8f __attribute__((ext_vector_type(8)));

#if defined(__AMDGCN__) && __has_builtin(__builtin_amdgcn_wmma_f32_16x16x4_f32)
#define HAVE_WMMA_F32X4 1
#else
#define HAVE_WMMA_F32X4 0
#endif

// ---------------------------------------------------------------------------
// Kernel 0: transpose/pad w2 [100,64,3,3] -> w2t [112, 576] with K ordered as
// (dy, dx, cc) so the conv GEMM's A/B fragments are contiguous 8B pairs.
// Rows 100..111 zero-padded (dead WMMA rows).
// ---------------------------------------------------------------------------
__global__ __launch_bounds__(256) void k0_prep_w2t(const float* __restrict__ w2,
                                                   float* __restrict__ w2t) {
  int idx = blockIdx.x * 256 + threadIdx.x;
  if (idx >= MCP * KK9) return;
  int row = idx / KK9, kidx = idx % KK9;
  int dy = kidx / 192, r = kidx % 192, dx = r >> 6, cc = r & 63;
  w2t[idx] = (row < MC) ? w2[row * KK9 + cc * 9 + dy * 3 + dx] : 0.0f;
}

// ---------------------------------------------------------------------------
// Kernel 1: 1x1 compressor as WMMA GEMM: comp = w1[64x256] * X[256 x N*H*W]
// Output stored NHWC: comp[n, hw, cc]  (friendly for the 3x3 conv B gather).
// One wave = one 16x16 tile; K loop 256 in steps of 4 (V_WMMA_F32_16X16X4_F32).
// ---------------------------------------------------------------------------
__global__ __launch_bounds__(256) void k1_compress(const float* __restrict__ x,
                                                   const float* __restrict__ w1,
                                                   const float* __restrict__ b1,
                                                   float* __restrict__ comp) {
#if HAVE_WMMA_F32X4
  int wave = blockIdx.x * 8 + (threadIdx.x >> 5);
  int lane = threadIdx.x & 31;
  int hl = lane >> 4, lid = lane & 15;
  int mt = wave & 3;            // 4 M tiles (64 rows)
  int ct = wave >> 2;           // 1024 N tiles (16384 cols)
  int n = ct >> 8;
  int hw0 = (ct & 255) << 4;    // 16 consecutive hw positions, never cross image
  const float* xb = x + n * (CI * HWSZ) + hw0 + lid;
  const float* ap = w1 + (mt * 16 + lid) * CI + 2 * hl;  // 8B-aligned pairs
  v8f acc = {};
  for (int k = 0; k < CI; k += 4) {
    int ka = k + 2 * hl;        // lanes 0-15: K=k,k+1 ; lanes 16-31: K=k+2,k+3
    v2f a = *(const v2f*)(ap + k);             // one b64 load
    v2f b;
    b.x = xb[ka * HWSZ];                       // coalesced: lanes adjacent hw
    b.y = xb[(ka + 1) * HWSZ];
    acc = __builtin_amdgcn_wmma_f32_16x16x4_f32(false, a, false, b,
                                                (short)0, acc, false, false);
  }
  int col = hw0 + lid;
#pragma unroll
  for (int r = 0; r < 8; ++r) {
    int row = mt * 16 + r + 8 * hl;            // C/D layout: M = r + 8*half
    comp[(n * HWSZ + col) * CCOMP + row] = acc[r] + b1[row];
  }
#else
  // VALU fallback (compile-safety only)
  int stride = gridDim.x * blockDim.x;
  for (int e = blockIdx.x * blockDim.x + threadIdx.x; e < CCOMP * NN * HWSZ; e += stride) {
    int row = e % CCOMP, col = e / CCOMP;
    int n = col / HWSZ, hw = col % HWSZ;
    float s = b1[row];
    for (int k = 0; k < CI; ++k) s += w1[row * CI + k] * x[(n * CI + k) * HWSZ + hw];
    comp[col * CCOMP + row] = s;
  }
#endif
}

// ---------------------------------------------------------------------------
// Kernel 2: 3x3 encoder as WMMA GEMM (im2col, K = 9*64 = 576) + fused
// pixel-shuffle-aware softmax over the 25 taps.
// One block per (n, h) row: 28 waves = 7 M-tiles (112 padded rows) x 4 pixel
// tiles (64 pixels). comp read from L2 (4 MB total, 192 MB L2). K loop is
// dy/dx/cc nests: incremental pointers, contiguous b64 fragment loads.
// ---------------------------------------------------------------------------
__global__ __launch_bounds__(896) void k2_mask_softmax(const float* __restrict__ comp,
                                                       const float* __restrict__ w2t,
                                                       const float* __restrict__ b2,
                                                       float* __restrict__ mws) {
  __shared__ float lds_mask[MCP * WW];        // 112*64*4B = 28 KB
  int n = blockIdx.x >> 6;
  int h = blockIdx.x & 63;
  int tid = threadIdx.x;
#if HAVE_WMMA_F32X4
  int wave = tid >> 5, lane = tid & 31, hl = lane >> 4, lid = lane & 15;
  int mt = wave >> 2;                          // 0..6
  int w0 = (wave & 3) << 4;                    // pixel tile base
  int wcol = w0 + lid;
  const float* abase = w2t + (mt * 16 + lid) * KK9 + 2 * hl;
  const float* cbase = comp + (size_t)n * HH * WW * CCOMP + 2 * hl;
  v8f acc = {};
  for (int dy = 0; dy < 3; ++dy) {
    int hrow = h + dy - 1;
    bool rowok = (unsigned)hrow < (unsigned)HH;
    for (int dx = 0; dx < 3; ++dx) {
      int wc = wcol + dx - 1;
      bool ok = rowok && ((unsigned)wc < (unsigned)WW);
      const float* bp = cbase + (hrow * WW + wc) * CCOMP;  // per-lane pixel
      const float* ap = abase + (dy * 3 + dx) * 64;
#pragma unroll
      for (int cc = 0; cc < 64; cc += 4) {
        v2f a = *(const v2f*)(ap + cc);        // contiguous (k0 transpose)
        v2f b = {0.0f, 0.0f};
        if (ok) b = *(const v2f*)(bp + cc);    // NHWC: cc contiguous
        acc = __builtin_amdgcn_wmma_f32_16x16x4_f32(false, a, false, b,
                                                    (short)0, acc, false, false);
      }
    }
  }
#pragma unroll
  for (int r = 0; r < 8; ++r) {
    int row = mt * 16 + r + 8 * hl;
    float v = acc[r] + ((row < MC) ? b2[row] : 0.0f);
    lds_mask[row * WW + w0 + lid] = v;         // lanes -> adjacent banks
  }
#else
  for (int e = tid; e < MCP * WW; e += 896) {
    int row = e >> 6, pix = e & 63;
    float s = (row < MC) ? b2[row] : 0.0f;
    if (row < MC) {
      for (int kidx = 0; kidx < KK9; ++kidx) {
        int dy = kidx / 192, r0 = kidx % 192, dx = r0 >> 6, cc = r0 & 63;
        int hrow = h + dy - 1, wc = pix + dx - 1;
        if ((unsigned)hrow < (unsigned)HH && (unsigned)wc < (unsigned)WW)
          s += w2t[row * KK9 + kidx] * comp[((n * HH + hrow) * WW + wc) * CCOMP + cc];
      }
    }
    lds_mask[row * WW + pix] = s;
  }
#endif
  __syncthreads();
  // pixel-shuffle + softmax: channel c2 = kk*4 + a*2 + b; softmax over kk per (pix, s)
  if (tid < 256) {
    int pix = tid & 63, s = tid >> 6;
    float v[KTAPS];
    float mx = -1e30f;
#pragma unroll
    for (int kk = 0; kk < KTAPS; ++kk) {
      v[kk] = lds_mask[(kk * 4 + s) * WW + pix];
      mx = fmaxf(mx, v[kk]);
    }
    float sum = 0.0f;
#pragma unroll
    for (int kk = 0; kk < KTAPS; ++kk) { v[kk] = __expf(v[kk] - mx); sum += v[kk]; }
    float inv = 1.0f / sum;
    float* mp = mws + ((n * HH + h) * WW + pix) * MCP + s * 28;
#pragma unroll
    for (int kk = 0; kk < KTAPS; ++kk) mp[kk] = v[kk] * inv;
  }
}

// ---------------------------------------------------------------------------
// Kernel 3: content-aware reassembly. Streaming kernel (output-write bound:
// 67 MB of the ~100 MB total traffic -> ~4.3 us roofline floor). Masks staged
// in LDS (odd stride 113 -> conflict-free), x taps served from L2 (17 MB of x
// vs 192 MB L2, ~25x reuse), next channel group's rows prefetched
// (global_prefetch_b8). Each thread writes two coalesced float2 (b64) stores.
// ---------------------------------------------------------------------------
__global__ __launch_bounds__(256) void k3_reassemble(const float* __restrict__ x,
                                                     const float* __restrict__ mws,
                                                     float* __restrict__ out) {
  __shared__ float lds_m[16 * 113];            // 7.2 KB
  int b = blockIdx.x;                          // 4*64*4 = 1024 blocks
  int wt = b & 3;
  int h = (b >> 2) & 63;
  int n = b >> 8;
  int w0 = wt << 4;
  int tid = threadIdx.x;
  int base = ((n * HH + h) * WW + w0) * MCP;
  for (int idx = tid; idx < 16 * MCP; idx += 256)
    lds_m[(idx / MCP) * 113 + (idx % MCP)] = mws[base + idx];  // contiguous loads
  __syncthreads();

  int pix = tid & 15, ci = tid >> 4;
  int wcen = w0 + pix;
  for (int cg = 0; cg < CI; cg += 16) {
    int c = cg + ci;
    const float* xp = x + (size_t)(n * CI + c) * HWSZ;

    // Prefetch the 5 rows of this thread's *next* channel toward L2/L0 while
    // the 25-tap reduction below runs (uniform branches; stays in-bounds).
    if (cg + 16 < CI) {
      const float* xn = xp + 16 * HWSZ;
#pragma unroll
      for (int dyp = -2; dyp <= 2; ++dyp) {
        int hyp = h + dyp;
        if ((unsigned)hyp < (unsigned)HH)
          __builtin_prefetch(xn + hyp * WW + wcen, 0, 3);
      }
    }

    float acc0 = 0.f, acc1 = 0.f, acc2 = 0.f, acc3 = 0.f;
#pragma unroll
    for (int kk = 0; kk < KTAPS; ++kk) {
      int dy = kk / 5 - 2, dx = kk % 5 - 2;
      int hy = h + dy, wx = wcen + dx;
      float xv = ((unsigned)hy < (unsigned)HH && (unsigned)wx < (unsigned)WW)
                     ? xp[hy * WW + wx] : 0.0f;  // lanes share hy -> coalesced
      const float* mp = &lds_m[pix * 113 + kk];
      acc0 += xv * mp[0];        // s=0 -> (a=0,b=0)
      acc1 += xv * mp[28];       // s=1 -> (a=0,b=1)
      acc2 += xv * mp[56];       // s=2 -> (a=1,b=0)
      acc3 += xv * mp[84];       // s=3 -> (a=1,b=1)
    }
    float* op = out + ((size_t)(n * CI + c) * HO + 2 * h) * WO + 2 * wcen;
    *(v2f*)op = (v2f){acc0, acc1};             // coalesced b64 stores
    *(v2f*)(op + WO) = (v2f){acc2, acc3};
  }
}

// ---------------------------------------------------------------------------
extern "C" void kernel_launch(void* const* d_in, const int* in_sizes, int n_in,
                              void* d_out, int out_size, void* d_ws, size_t ws_size,
                              hipStream_t stream) {
  const float* x  = (const float*)d_in[0];
  const float* w1 = (const float*)d_in[1];
  const float* b1 = (const float*)d_in[2];
  const float* w2 = (const float*)d_in[3];
  const float* b2 = (const float*)d_in[4];
  float* out = (float*)d_out;

  // Workspace layout (floats): comp NHWC | w2t transposed | normalized masks
  float* comp = (float*)d_ws;                  // 4*4096*64   = 1,048,576
  float* w2t  = comp + NN * HWSZ * CCOMP;      // 112*576     =    64,512
  float* mws  = w2t + MCP * KK9;               // 4*4096*112  = 1,835,008
                                               // total ~11.8 MB

  k0_prep_w2t<<<(MCP * KK9 + 255) / 256, 256, 0, stream>>>(w2, w2t);
  k1_compress<<<(4 * (NN * HWSZ / 16)) / 8, 256, 0, stream>>>(x, w1, b1, comp);
  k2_mask_softmax<<<NN * HH, 896, 0, stream>>>(comp, w2t, b2, mws);
  k3_reassemble<<<NN * HH * (WW / 16), 256, 0, stream>>>(x, mws, out);
}